// Sinkhorn_18588618457438
// MI455X (gfx1250) — compile-verified
//
#include <hip/hip_runtime.h>

// Sinkhorn (10 alternating col/row normalizations) on 8192x8192 f32.
// Factored form: state is always diag(r) * (s+eps) * diag(c), so each
// iteration is one streaming reduction over the matrix; output is one
// elementwise pass. HBM traffic ~3.1GB -> ~132us at 23.3 TB/s.

#define SK_N 8192                    // rows == cols
#define SK_CHUNKS 32                 // row chunks for column-sum partials
#define SK_ROWS (SK_N / SK_CHUNKS)   // 256 rows per chunk
static constexpr float kEps = 1e-4f;

typedef __attribute__((ext_vector_type(2))) float v2f;
typedef __attribute__((ext_vector_type(8))) float v8f;

// r[i] = 1, c[j] = 1
__global__ void sk_init_ones(float* __restrict__ r, float* __restrict__ c) {
    int i = blockIdx.x * blockDim.x + threadIdx.x;
    if (i < SK_N) { r[i] = 1.0f; c[i] = 1.0f; }
}

// Stage 1 of weighted column sums via V_WMMA_F32_16X16X4_F32.
// partial[chunk][j] = sum over 256 rows of (s[i][j]+eps) * r[i].
//
// The r-weighting is applied on the VALU into the B operand, and A is a
// constant all-ones 16x4 matrix, so D[m][n] = sum_k B[k][n] is invariant to
// the K->(VGPR,half) register convention of either operand; only the
// documented C/D layout (VGPR0, lanes 0-15 = M=0, N=lane) is needed to
// extract the 16 column sums. Each wave owns four 16-column tiles (64 cols)
// with independent accumulators: the per-row r handling is amortized 4x and
// the wave reads a contiguous 256B per row.
__global__ void sk_colsum_wmma(const float* __restrict__ X,
                               const float* __restrict__ r,
                               float* __restrict__ partial) {
    const int lane = threadIdx.x & 31;
    const int wave = threadIdx.x >> 5;          // 0..7
    const int hi   = lane >> 4;                 // lanes 0-15 rows {k,k+1}; 16-31 rows {k+2,k+3}
    const int j0   = (blockIdx.x * 8 + wave) * 64 + (lane & 15);
    const int i0   = blockIdx.y * SK_ROWS;

    const float* __restrict__ rp = r + i0;
    const float* __restrict__ p  = X + (size_t)(i0 + 2 * hi) * SK_N + j0;

    v2f ones; ones[0] = 1.0f; ones[1] = 1.0f;   // constant A operand
    v8f acc0 = {}, acc1 = {}, acc2 = {}, acc3 = {};

    for (int k = 0; k < SK_ROWS; k += 4) {
        float r0 = rp[k + 0], r1 = rp[k + 1], r2 = rp[k + 2], r3 = rp[k + 3];
        float rv0 = hi ? r2 : r0;               // r for this lane-half's first row
        float rv1 = hi ? r3 : r1;               // r for this lane-half's second row
        __builtin_prefetch((const void*)(p + 16 * (size_t)SK_N), 0, 0);

        v2f B0, B1, B2, B3;
        B0[0] = (p[0]          + kEps) * rv0;  B0[1] = (p[SK_N]      + kEps) * rv1;
        B1[0] = (p[16]         + kEps) * rv0;  B1[1] = (p[SK_N + 16] + kEps) * rv1;
        B2[0] = (p[32]         + kEps) * rv0;  B2[1] = (p[SK_N + 32] + kEps) * rv1;
        B3[0] = (p[48]         + kEps) * rv0;  B3[1] = (p[SK_N + 48] + kEps) * rv1;

        acc0 = __builtin_amdgcn_wmma_f32_16x16x4_f32(
            false, ones, false, B0, (short)0, acc0, false, false);
        acc1 = __builtin_amdgcn_wmma_f32_16x16x4_f32(
            false, ones, false, B1, (short)0, acc1, false, false);
        acc2 = __builtin_amdgcn_wmma_f32_16x16x4_f32(
            false, ones, false, B2, (short)0, acc2, false, false);
        acc3 = __builtin_amdgcn_wmma_f32_16x16x4_f32(
            false, ones, false, B3, (short)0, acc3, false, false);

        p += 4 * (size_t)SK_N;
    }

    if (lane < 16) {
        size_t base = (size_t)blockIdx.y * SK_N + j0;
        partial[base +  0] = acc0[0];
        partial[base + 16] = acc1[0];
        partial[base + 32] = acc2[0];
        partial[base + 48] = acc3[0];
    }
}

// Stage 2: c[j] = 1 / sum_chunks partial[chunk][j]  (fixed order, deterministic)
__global__ void sk_colsum_reduce(const float* __restrict__ partial,
                                 float* __restrict__ c) {
    int j = blockIdx.x * blockDim.x + threadIdx.x;
    float s = 0.0f;
    for (int k = 0; k < SK_CHUNKS; ++k) s += partial[(size_t)k * SK_N + j];
    c[j] = 1.0f / s;
}

// r[row] = 1 / sum_j (s[row][j]+eps) * c[j]   — one block per row, coalesced
// float4 dot + fixed-order LDS tree reduction (bitwise deterministic).
__global__ void sk_rowsum(const float* __restrict__ X,
                          const float* __restrict__ c,
                          float* __restrict__ r) {
    __shared__ float red[256];
    const int row = blockIdx.x;
    const float4* __restrict__ xr = (const float4*)(X + (size_t)row * SK_N);
    const float4* __restrict__ cc = (const float4*)c;
    float acc = 0.0f;
    for (int j = threadIdx.x; j < SK_N / 4; j += 256) {
        float4 x  = xr[j];
        float4 cv = cc[j];
        acc += (x.x + kEps) * cv.x + (x.y + kEps) * cv.y +
               (x.z + kEps) * cv.z + (x.w + kEps) * cv.w;
    }
    red[threadIdx.x] = acc;
    __syncthreads();
    for (int s = 128; s > 0; s >>= 1) {
        if (threadIdx.x < s) red[threadIdx.x] += red[threadIdx.x + s];
        __syncthreads();
    }
    if (threadIdx.x == 0) r[row] = 1.0f / red[0];
}

// out[i][j] = (s[i][j]+eps) * r[i] * c[j]  — float4 streaming pass.
// 2048 float4 per row, 256 threads/block -> each block stays in one row.
__global__ void sk_finalize(const float* __restrict__ X,
                            const float* __restrict__ r,
                            const float* __restrict__ c,
                            float* __restrict__ out) {
    size_t idx = (size_t)blockIdx.x * blockDim.x + threadIdx.x;  // over N*N/4
    int row = (int)(idx >> 11);          // / (N/4) = /2048
    int j4  = (int)(idx & 2047);
    float4 x  = ((const float4*)X)[idx];
    float4 cv = ((const float4*)c)[j4];
    float  rv = r[row];
    float4 o;
    o.x = (x.x + kEps) * rv * cv.x;
    o.y = (x.y + kEps) * rv * cv.y;
    o.z = (x.z + kEps) * rv * cv.z;
    o.w = (x.w + kEps) * rv * cv.w;
    ((float4*)out)[idx] = o;
}

extern "C" void kernel_launch(void* const* d_in, const int* in_sizes, int n_in,
                              void* d_out, int out_size, void* d_ws, size_t ws_size,
                              hipStream_t stream) {
    (void)in_sizes; (void)n_in; (void)out_size; (void)ws_size;
    const float* X = (const float*)d_in[0];     // [1, 8192, 8192] f32
    float* out = (float*)d_out;                 // [8192, 8192] f32

    // workspace: r (8192 f32) | c (8192 f32) | partial (32 * 8192 f32) ≈ 1.1 MB
    float* r = (float*)d_ws;
    float* c = r + SK_N;
    float* partial = c + SK_N;

    sk_init_ones<<<SK_N / 256, 256, 0, stream>>>(r, c);

    for (int it = 0; it < 10; ++it) {
        if ((it & 1) == 0) {
            // column normalization: c = 1 / (X^T r)
            sk_colsum_wmma<<<dim3(SK_N / 512, SK_CHUNKS), 256, 0, stream>>>(X, r, partial);
            sk_colsum_reduce<<<SK_N / 256, 256, 0, stream>>>(partial, c);
        } else {
            // row normalization: r = 1 / (X c)
            sk_rowsum<<<SK_N, 256, 0, stream>>>(X, c, r);
        }
    }

    sk_finalize<<<(SK_N / 4) * (SK_N / 256), 256, 0, stream>>>(X, r, c, out);
}